// SoftCrossEntropyLossByNegSampling_31327491457332
// MI455X (gfx1250) — compile-verified
//
#include <hip/hip_runtime.h>
#include <stdint.h>
#include <stddef.h>

// ---------------------------------------------------------------------------
// MI455X (gfx1250, wave32) implementation of negative-sampling soft CE loss.
// Bandwidth-bound: one streaming pass over output+target (400 MB -> ~17us at
// 23.3 TB/s). Negative sampling is regenerated from a counter-based hash
// (no extra HBM traffic); per-row top-k via Poisson-thresholded candidate
// collection in LDS + exact rank counting. Wave reductions use
// V_WMMA_F32_16X16X4_F32 with an all-ones B matrix (f32-exact row sums).
// ---------------------------------------------------------------------------

typedef float v2f __attribute__((ext_vector_type(2)));
typedef float v8f __attribute__((ext_vector_type(8)));

// Full-precision 32-lane wave sum using one WMMA:
// A[m][k] layout (16x4 f32): lanes 0-15 hold (M=lane, K=0..1) in 2 VGPRs,
// lanes 16-31 hold (M=lane-16, K=2..3). With B == ones(4x16):
//   D[m][n] = A[m][0]+A[m][2] = p_m + p_{m+16}   (for a = {p_lane, 0})
// Each lane then sums its 8 D VGPRs (rows 0-7 or 8-15 of D) and one
// shfl_xor(16) completes the 32-lane sum, broadcast to every lane.
__device__ __forceinline__ float wave_sum32(float v) {
    v2f a; a[0] = v;    a[1] = 0.0f;
    v2f b; b[0] = 1.0f; b[1] = 1.0f;
    v8f c = {};
    c = __builtin_amdgcn_wmma_f32_16x16x4_f32(false, a, false, b,
                                              (short)0, c, false, false);
    float s = ((c[0] + c[1]) + (c[2] + c[3])) + ((c[4] + c[5]) + (c[6] + c[7]));
    s += __shfl_xor(s, 16, 32);
    return s;
}

// Block-wide sum (blockDim.x must be a multiple of 32, <= 256).
__device__ __forceinline__ float block_sum(float v, float* smem8) {
    const int lane = threadIdx.x & 31;
    const int wid  = threadIdx.x >> 5;
    const int nw   = blockDim.x >> 5;
    float s = wave_sum32(v);
    if (lane == 0) smem8[wid] = s;
    __syncthreads();
    float tot = 0.0f;
    for (int w = 0; w < nw; ++w) tot += smem8[w];
    __syncthreads();
    return tot;
}

// Numerically stable log(sigmoid(x)) = min(x,0) - log1p(exp(-|x|)).
__device__ __forceinline__ float log_sigmoid(float x) {
    float e = __expf(-fabsf(x));
    return fminf(x, 0.0f) - __logf(1.0f + e);
}

// Stateless counter-based Gumbel(0,1): pure function of (row, col).
__device__ __forceinline__ float gumbel01(uint32_t i, uint32_t j) {
    uint32_t h = i * 0x9E3779B1u + j * 0x85EBCA77u;
    h ^= h >> 16; h *= 0x7FEB352Du;
    h ^= h >> 15; h *= 0x846CA68Bu;
    h ^= h >> 16;
    // u in (0,1), never exactly 0 or 1
    float u = (float)(h >> 8) * (1.0f / 16777216.0f) + (0.5f / 16777216.0f);
    return -__logf(-__logf(u));
}

// ---------------------------------------------------------------------------
// Kernel 0: log of noise probs (tiny; keeps pass B free of extra logs).
// ---------------------------------------------------------------------------
__global__ void k_lognoise(const float* __restrict__ p,
                           float* __restrict__ lp, int n) {
    int j = blockIdx.x * blockDim.x + threadIdx.x;
    if (j < n) lp[j] = __logf(p[j]);
}

// ---------------------------------------------------------------------------
// Pass A: one block per row. Streams output+target with b128 loads.
// Computes s1[i] = sum target*log_sigmoid(output) and k[i] = min(npos*nns, N).
// ---------------------------------------------------------------------------
__global__ void k_pos_pass(const float* __restrict__ out,
                           const float* __restrict__ tgt,
                           const int*   __restrict__ nns_p,
                           int N,
                           float* __restrict__ rowS1,
                           int*   __restrict__ rowK) {
    __shared__ float sm8[8];
    __shared__ int   sh_npos;
    const int row = blockIdx.x;
    const int tid = threadIdx.x;
    if (tid == 0) sh_npos = 0;
    __syncthreads();

    const float* o = out + (size_t)row * (size_t)N;
    const float* t = tgt + (size_t)row * (size_t)N;

    float s1 = 0.0f;
    int npos = 0;

    if ((N & 3) == 0) {
        const float4* o4 = (const float4*)o;
        const float4* t4 = (const float4*)t;
        const int n4 = N >> 2;
        for (int j = tid; j < n4; j += blockDim.x) {
            float4 ov = o4[j];
            float4 tv = t4[j];
            // target is ~0.02% dense: skip transcendentals on the fast path
            if (tv.x != 0.0f) { s1 += tv.x * log_sigmoid(ov.x); ++npos; }
            if (tv.y != 0.0f) { s1 += tv.y * log_sigmoid(ov.y); ++npos; }
            if (tv.z != 0.0f) { s1 += tv.z * log_sigmoid(ov.z); ++npos; }
            if (tv.w != 0.0f) { s1 += tv.w * log_sigmoid(ov.w); ++npos; }
        }
    } else {
        for (int j = tid; j < N; j += blockDim.x) {
            float tv = t[j];
            if (tv != 0.0f) { s1 += tv * log_sigmoid(o[j]); ++npos; }
        }
    }

    atomicAdd(&sh_npos, npos);
    float s1tot = block_sum(s1, sm8);   // barrier inside makes sh_npos visible
    if (tid == 0) {
        rowS1[row] = s1tot;
        long long k = (long long)sh_npos * (long long)nns_p[0];
        rowK[row] = (int)(k < (long long)N ? k : (long long)N);
    }
}

// ---------------------------------------------------------------------------
// Pass B: one block per row. Gumbel-top-k without materializing scores.
// Count of scores above t is Poisson(e^-t) (probs normalized), so threshold
// at -log(~6k) gathers ~6k candidates into LDS; exact top-k by rank counting.
// Only the selected ~k elements of `output` are gathered from memory.
// ---------------------------------------------------------------------------
#define NEG_CAP 4096

__global__ void k_neg_pass(const float* __restrict__ out,
                           const float* __restrict__ logp,
                           const int*   __restrict__ rowK,
                           int N,
                           float* __restrict__ rowS2) {
    __shared__ float sm8[8];
    __shared__ int   sh_cnt;
    __shared__ float sc[NEG_CAP];
    __shared__ int   si[NEG_CAP];

    const int row = blockIdx.x;
    const int tid = threadIdx.x;
    const int k   = rowK[row];           // uniform across block

    float s2 = 0.0f;
    if (k > 0) {
        float lam = (float)(6 * k + 128);
        if (lam > 3500.0f) lam = 3500.0f;
        float thr = -__logf(lam);

        int c = 0;
        for (int attempt = 0; attempt < 8; ++attempt) {
            if (tid == 0) sh_cnt = 0;
            __syncthreads();
            for (int j = tid; j < N; j += blockDim.x) {
                float s = logp[j] + gumbel01((uint32_t)row, (uint32_t)j);
                if (s > thr) {
                    int p = atomicAdd(&sh_cnt, 1);
                    if (p < NEG_CAP) { sc[p] = s; si[p] = j; }
                }
            }
            __syncthreads();
            c = sh_cnt;                  // uniform
            __syncthreads();             // all reads done before next reset
            if (c >= k && c <= NEG_CAP) break;
            // overflow: halve expected count; underflow: 4x it (deterministic)
            thr += (c > NEG_CAP) ? 0.69314718f : -1.38629436f;
        }

        const int cu = (c < NEG_CAP) ? c : NEG_CAP;
        const int kk = (k < cu) ? k : cu;
        const float* o = out + (size_t)row * (size_t)N;

        // Exact selection: rank p = #candidates strictly ahead of p.
        for (int p = tid; p < cu; p += blockDim.x) {
            const float sp = sc[p];
            int rank = 0;
            for (int q = 0; q < cu; ++q) {
                float sq = sc[q];
                rank += (sq > sp) ? 1 : ((sq == sp && q < p) ? 1 : 0);
            }
            if (rank < kk) s2 += log_sigmoid(-o[si[p]]);
        }
    }
    __syncthreads();
    float tot = block_sum(s2, sm8);
    if (tid == 0) rowS2[row] = tot;
}

// ---------------------------------------------------------------------------
// Final: loss = -mean(s1 + s2)
// ---------------------------------------------------------------------------
__global__ void k_final(const float* __restrict__ rowS1,
                        const float* __restrict__ rowS2,
                        int Mrows, float* __restrict__ outp) {
    __shared__ float sm8[8];
    float v = 0.0f;
    for (int i = threadIdx.x; i < Mrows; i += blockDim.x)
        v += rowS1[i] + rowS2[i];
    float tot = block_sum(v, sm8);
    if (threadIdx.x == 0) outp[0] = -tot / (float)Mrows;
}

// ---------------------------------------------------------------------------
extern "C" void kernel_launch(void* const* d_in, const int* in_sizes, int n_in,
                              void* d_out, int out_size, void* d_ws, size_t ws_size,
                              hipStream_t stream) {
    const float* output = (const float*)d_in[0];
    const float* target = (const float*)d_in[1];
    const float* noise  = (const float*)d_in[2];
    const int*   nns    = (const int*)d_in[3];

    const int N = in_sizes[2];               // vocab size
    const int M = in_sizes[0] / N;           // rows

    // Workspace layout: rowS1[M] | rowS2[M] | rowK[M] | logNoise[N]
    float* rowS1 = (float*)d_ws;
    float* rowS2 = rowS1 + M;
    int*   rowK  = (int*)(rowS2 + M);
    float* logN  = (float*)(rowK + M);

    k_lognoise<<<(N + 255) / 256, 256, 0, stream>>>(noise, logN, N);
    k_pos_pass<<<M, 256, 0, stream>>>(output, target, nns, N, rowS1, rowK);
    k_neg_pass<<<M, 256, 0, stream>>>(output, logN, rowK, N, rowS2);
    k_final<<<1, 256, 0, stream>>>(rowS1, rowS2, M, (float*)d_out);
}